// Sample_72885595013524
// MI455X (gfx1250) — compile-verified
//
#include <hip/hip_runtime.h>

// Radon projection for MI455X (gfx1250, wave32).
// out[ch][h][x] = sgn(ch) * sum_y bilinear(vol_ch[:, h, :], ys(y,x), xs(y,x))
//   ys = c*(y-cy) - s*(x-cx) + cy ;  xs = s*(y-cy) + c*(x-cx) + cx
// The 384-deep ray accumulation runs on the matrix pipe:
// V_WMMA_F32_16X16X4_F32 with A = ones(16x4) gives D[m,n] = sum_k B[k,n] + C,
// exact in f32, freeing the VALU for interpolation math (XDL co-execution).
//
// The ray loop is split into guarded-prologue / fast-middle / guarded-epilogue.
// The interior t-interval is computed once per wave (analytically, from the
// wave's extreme columns) and held in SGPRs via readfirstlane, so all loops
// are scalar-controlled and EXEC stays all-ones at every WMMA.

typedef __attribute__((ext_vector_type(2))) float v2f;
typedef __attribute__((ext_vector_type(8))) float v8f;

#define NDIM   384
#define NDIM2  (384u * 384u)     // y-row stride in elements
#define CENTER 191.5f            // (384-1)/2

__device__ __forceinline__ int clampi(int v, int lo, int hi) {
    return v < lo ? lo : (v > hi ? hi : v);
}

// Interior sample: all four taps guaranteed inside the slice.
__device__ __forceinline__ float sample_fast(const float* __restrict__ img,
                                             unsigned hbase,
                                             float ys, float xs) {
    const float fy = floorf(ys), fx = floorf(xs);
    const float wy = ys - fy,   wx = xs - fx;
    const unsigned r0 = (unsigned)(int)fy * NDIM2 + hbase + (unsigned)(int)fx;
    const float v00 = img[r0];
    const float v01 = img[r0 + 1];
    const float v10 = img[r0 + NDIM2];
    const float v11 = img[r0 + NDIM2 + 1];
    const float top = fmaf(wx, v01 - v00, v00);
    const float bot = fmaf(wx, v11 - v10, v10);
    return fmaf(wy, bot - top, top);
}

// Boundary sample: branchless, zeros outside [0,384)^2 via masked weights.
__device__ __forceinline__ float sample_guarded(const float* __restrict__ img,
                                                unsigned hbase,
                                                float ys, float xs) {
    const float fy = floorf(ys), fx = floorf(xs);
    const float wy = ys - fy,   wx = xs - fx;
    const int y0 = (int)fy,     x0 = (int)fx;

    const bool iy0 = (unsigned)y0       < (unsigned)NDIM;
    const bool iy1 = (unsigned)(y0 + 1) < (unsigned)NDIM;
    const bool ix0 = (unsigned)x0       < (unsigned)NDIM;
    const bool ix1 = (unsigned)(x0 + 1) < (unsigned)NDIM;

    const unsigned r0 = (unsigned)clampi(y0,     0, NDIM - 1) * NDIM2 + hbase;
    const unsigned r1 = (unsigned)clampi(y0 + 1, 0, NDIM - 1) * NDIM2 + hbase;
    const unsigned c0 = (unsigned)clampi(x0,     0, NDIM - 1);
    const unsigned c1 = (unsigned)clampi(x0 + 1, 0, NDIM - 1);

    const float v00 = img[r0 + c0];
    const float v01 = img[r0 + c1];
    const float v10 = img[r1 + c0];
    const float v11 = img[r1 + c1];

    const float ay = 1.0f - wy, ax = 1.0f - wx;
    const float w00 = (iy0 && ix0) ? ay * ax : 0.0f;
    const float w01 = (iy0 && ix1) ? ay * wx : 0.0f;
    const float w10 = (iy1 && ix0) ? wy * ax : 0.0f;
    const float w11 = (iy1 && ix1) ? wy * wx : 0.0f;

    float acc = w00 * v00;
    acc = fmaf(w01, v01, acc);
    acc = fmaf(w10, v10, acc);
    acc = fmaf(w11, v11, acc);
    return acc;
}

__global__ __launch_bounds__(256) void radon_wmma_kernel(
    const float* __restrict__ delta,
    const float* __restrict__ beta,
    const float* __restrict__ angle_ptr,
    float* __restrict__ out)
{
    const int lane  = threadIdx.x & 31;
    const int wave  = threadIdx.x >> 5;
    const int ncol  = lane & 15;   // output column within the wave's 16-wide tile
    const int khalf = lane >> 4;   // which k-pair {0,1} / {2,3} of the 4-deep B tile

    const int xbase = blockIdx.x * 128 + wave * 16;      // wave's first column
    const int x  = xbase + ncol;                         // output column (0..383)
    const int h  = blockIdx.y;                           // slice index
    const int ch = blockIdx.z;                           // 0 = Re(delta), 1 = Im(-beta)

    const float* __restrict__ img = (ch == 0) ? delta : beta;
    const float sgn = (ch == 0) ? 1.0f : -1.0f;
    const unsigned hbase = (unsigned)h * (unsigned)NDIM;

    // angle in [0, pi): hardware sin/cos (v_sin_f32/v_cos_f32), no Payne-Hanek.
    const float ang = angle_ptr[0];
    const float c = __cosf(ang);
    const float s = __sinf(ang);

    const float Xr = (float)x - CENTER;
    // ys(t) = c*(t-CENTER) + ys_col ; xs(t) = s*(t-CENTER) + xs_col
    const float ys_col = fmaf(-s, Xr, CENTER);
    const float xs_col = fmaf( c, Xr, CENTER);

    // ---- Wave-uniform interior t-interval [tlo, thi] (computed once). ----
    // Envelope of ys_col/xs_col over the wave's 16 columns (monotonic in Xr),
    // widened by 0.05 for incremental-update drift.
    {
        // scope only for readability; results land in fs/fe below
    }
    const float Xlo = (float)xbase - CENTER;
    const float Xhi = (float)(xbase + 15) - CENTER;
    const float yc0 = fmaf(-s, Xlo, CENTER), yc1 = fmaf(-s, Xhi, CENTER);
    const float xc0 = fmaf( c, Xlo, CENTER), xc1 = fmaf( c, Xhi, CENTER);
    const float ybl = fminf(yc0, yc1) - 0.05f, ybh = fmaxf(yc0, yc1) + 0.05f;
    const float xbl = fminf(xc0, xc1) - 0.05f, xbh = fmaxf(xc0, xc1) + 0.05f;

    const float LO = 0.25f, HI = 382.25f;   // safe interior coord range
    float tlo = 0.0f, thi = 383.0f;
    // constraint: a*(t-CENTER) + b in [LO,HI] for all b in [bl,bh]
    {
        const float a = c, bl = ybl, bh = ybh;
        if (a > 0.0f) {
            tlo = fmaxf(tlo, CENTER + (LO - bl) / a);
            thi = fminf(thi, CENTER + (HI - bh) / a);
        } else if (a < 0.0f) {
            tlo = fmaxf(tlo, CENTER + (HI - bh) / a);
            thi = fminf(thi, CENTER + (LO - bl) / a);
        } else if (!(bl >= LO && bh <= HI)) {
            tlo = 1.0f; thi = 0.0f;  // empty
        }
    }
    {
        const float a = s, bl = xbl, bh = xbh;
        if (a > 0.0f) {
            tlo = fmaxf(tlo, CENTER + (LO - bl) / a);
            thi = fminf(thi, CENTER + (HI - bh) / a);
        } else if (a < 0.0f) {
            tlo = fmaxf(tlo, CENTER + (HI - bh) / a);
            thi = fminf(thi, CENTER + (LO - bl) / a);
        } else if (!(bl >= LO && bh <= HI)) {
            tlo = 1.0f; thi = 0.0f;  // empty
        }
    }

    int fs, fe;  // fast region: blocks yb in [fs, fe), multiples of 4
    if (tlo <= thi) {
        const int st = ((int)ceilf(tlo) + 3) & ~3;          // first yb >= tlo
        const int en = (((int)floorf(thi) - 3) & ~3) + 4;   // past last yb with yb+3 <= thi
        fs = clampi(st, 0, NDIM);
        fe = clampi(en, fs, NDIM);
    } else {
        fs = NDIM; fe = NDIM;
    }
    // Values are identical across the wave; pin them to SGPRs so all three
    // loops are scalar-controlled (EXEC stays all-ones -> WMMA legal).
    fs = __builtin_amdgcn_readfirstlane(fs);
    fe = __builtin_amdgcn_readfirstlane(fe);

    // This lane's first ray step is t = khalf*2; advance incrementally by
    // (c, s) per step and (4c, 4s) per WMMA block.
    const float Yr0 = (float)(khalf * 2) - CENTER;
    float ysA = fmaf(c, Yr0, ys_col);
    float xsA = fmaf(s, Yr0, xs_col);
    const float dys4 = 4.0f * c, dxs4 = 4.0f * s;

    v8f acc = {};                               // 16x16 f32 accumulator
    v2f ones; ones[0] = 1.0f; ones[1] = 1.0f;   // A = ones(16x4)

    int yb = 0;
    // --- guarded prologue ---
    for (; yb < fs; yb += 4) {
        v2f b;
        b[0] = sample_guarded(img, hbase, ysA,     xsA);
        b[1] = sample_guarded(img, hbase, ysA + c, xsA + s);
        acc = __builtin_amdgcn_wmma_f32_16x16x4_f32(
            false, ones, false, b, (short)0, acc, false, false);
        ysA += dys4; xsA += dxs4;
    }
    // --- fast interior middle (no clamps, no masks, no tests) ---
    for (; yb < fe; yb += 4) {
        v2f b;
        b[0] = sample_fast(img, hbase, ysA,     xsA);
        b[1] = sample_fast(img, hbase, ysA + c, xsA + s);
        acc = __builtin_amdgcn_wmma_f32_16x16x4_f32(
            false, ones, false, b, (short)0, acc, false, false);
        ysA += dys4; xsA += dxs4;
    }
    // --- guarded epilogue ---
    for (; yb < NDIM; yb += 4) {
        v2f b;
        b[0] = sample_guarded(img, hbase, ysA,     xsA);
        b[1] = sample_guarded(img, hbase, ysA + c, xsA + s);
        acc = __builtin_amdgcn_wmma_f32_16x16x4_f32(
            false, ones, false, b, (short)0, acc, false, false);
        ysA += dys4; xsA += dxs4;
    }

    // C/D layout: VGPR0 = row M=0 for lanes 0-15 (N=lane). All rows of D are
    // identical (A rows identical), so lanes 0-15 hold the 16 column sums.
    if (lane < 16) {
        out[((size_t)ch * NDIM + h) * NDIM + x] = sgn * acc[0];
    }
}

extern "C" void kernel_launch(void* const* d_in, const int* in_sizes, int n_in,
                              void* d_out, int out_size, void* d_ws, size_t ws_size,
                              hipStream_t stream) {
    const float* delta = (const float*)d_in[0];
    const float* beta  = (const float*)d_in[1];
    const float* angle = (const float*)d_in[2];
    float* out = (float*)d_out;

    dim3 block(256, 1, 1);                 // 8 waves -> 128 output columns per block
    dim3 grid(NDIM / 128, NDIM, 2);        // (x-tiles, h, channel)
    hipLaunchKernelGGL(radon_wmma_kernel, grid, block, 0, stream,
                       delta, beta, angle, out);
}